// NoisyTopKGate_52750788329544
// MI455X (gfx1250) — compile-verified
//
#include <hip/hip_runtime.h>
#include <math.h>

// ---------------------------------------------------------------------------
// NoisyTopKGate for gfx1250 (MI455X).
//
// Two GEMMs [65536x1024]x[1024x64] via V_WMMA_F32_16X16X4_F32, with the
// Tensor Data Mover (tensor_load_to_lds) streaming double-buffered x/W
// chunks into LDS (270 KB of the WGP's 320 KB), TENSORcnt-pipelined against
// the WMMA stream. Softplus / noisy-H / top-2 / softmax epilogue. wave32.
// ---------------------------------------------------------------------------

typedef __attribute__((ext_vector_type(2))) float        v2f;
typedef __attribute__((ext_vector_type(4))) float        v4f;
typedef __attribute__((ext_vector_type(8))) float        v8f;
typedef __attribute__((ext_vector_type(4))) unsigned int v4u;
typedef __attribute__((ext_vector_type(4))) int          v4i;
typedef __attribute__((ext_vector_type(8))) int          v8i;

#define TOKENS      65536
#define MDIM        1024
#define NEXP        64
#define KCHUNK      128
#define NCHUNKS     (MDIM / KCHUNK)            // 8
#define LDSPAD      132                        // 128 + 4 pad floats per row
#define BLK_TOK     128                        // 8 waves * 16 tokens
#define X_FLOATS    (BLK_TOK * LDSPAD)         // 16896
#define W_FLOATS    (NEXP * LDSPAD)            // 8448
#define STAGE_FLOATS (X_FLOATS + 2 * W_FLOATS) // 33792 floats / stage
#define SMEM_BYTES  (2 * STAGE_FLOATS * 4)     // 270336 B double-buffered

#if defined(__has_builtin)
#if __has_builtin(__builtin_amdgcn_tensor_load_to_lds)
#define HAVE_TDM 1
#endif
#endif

extern __shared__ float smem[];

// Issue one TDM 2D tile load: tile_w x tile_h fp32 tile from a row-major
// [tensor_h x tensor_w] tensor (row stride `stride` elements) into LDS at
// byte offset lds_off, with HW padding of 4 DWORDs after every 128 DWORDs
// (pad_interval=6 -> 128 DW, pad_amount=3 -> 4 DW) => 132-float LDS rows.
__device__ __forceinline__ void tdm_load_tile_f32(unsigned lds_off,
                                                  const float* gptr,
                                                  unsigned tensor_w,
                                                  unsigned tensor_h,
                                                  unsigned stride,
                                                  unsigned tile_w,
                                                  unsigned tile_h)
{
#ifdef HAVE_TDM
    unsigned long long ga = (unsigned long long)(size_t)gptr;
    v4u g0;
    g0.x = 1u;                                             // count=1, user D#
    g0.y = lds_off;                                        // lds_addr (bytes)
    g0.z = (unsigned)ga;                                   // global_addr[31:0]
    g0.w = (unsigned)((ga >> 32) & 0x01FFFFFFu)            // global_addr[56:32]
         | (2u << 30);                                     // type=2 ("image")
    v8i g1;
    g1[0] = (int)((2u << 16)      // data_size = 4 bytes
                | (1u << 20)      // pad_enable
                | (6u << 22)      // pad_interval: 128 DWORDs
                | (3u << 25));    // pad_amount:   4 DWORDs
    g1[1] = (int)((tensor_w & 0xFFFFu) << 16);                           // dim0[15:0]
    g1[2] = (int)(((tensor_w >> 16) & 0xFFFFu) | ((tensor_h & 0xFFFFu) << 16));
    g1[3] = (int)(((tensor_h >> 16) & 0xFFFFu) | ((tile_w & 0xFFFFu) << 16));
    g1[4] = (int)(tile_h & 0xFFFFu);                                     // tile_dim1
    g1[5] = (int)stride;                                                 // dim0_stride[31:0]
    g1[6] = 0;
    g1[7] = 0;
    v4i z4 = (v4i)0;   // groups 2/3: 2D tensor, unused dims zero
    v8i z8 = (v8i)0;   // extra descriptor words (6-arg toolchain form)
    __builtin_amdgcn_tensor_load_to_lds(g0, g1, z4, z4, z8, 0);
#endif
}

__global__ __launch_bounds__(256, 1)
void noisy_topk_gate_kernel(const float* __restrict__ x,
                            const float* __restrict__ noise,
                            const float* __restrict__ Wg_w,
                            const float* __restrict__ Wg_b,
                            const float* __restrict__ Wn_w,
                            const float* __restrict__ Wn_b,
                            float* __restrict__ gates_out,
                            float* __restrict__ H_out,
                            float* __restrict__ idx_out,
                            float* __restrict__ ns_out,
                            float* __restrict__ lg_out)
{
    const int tid  = threadIdx.x;
    const int wave = tid >> 5;       // 0..7
    const int lane = tid & 31;       // wave32 lane
    const int l16  = lane & 15;
    const int hi   = lane >> 4;      // lane half selects K-pair (A/B frag layout)
    const int tok0 = blockIdx.x * BLK_TOK;

    float* stage0 = smem;
    float* stage1 = smem + STAGE_FLOATS;

    // Accumulators: 4 N-tiles (64 experts) x {gate, noise} GEMMs.
    v8f accG[4], accN[4];
#pragma unroll
    for (int nt = 0; nt < 4; ++nt) {
        accG[nt] = (v8f)0.0f;
        accN[nt] = (v8f)0.0f;
    }

    // Per-lane biases (expert = nt*16 + l16; same for every row in the lane).
    float bg[4], bn[4];
#pragma unroll
    for (int nt = 0; nt < 4; ++nt) {
        bg[nt] = Wg_b[nt * 16 + l16];
        bn[nt] = Wn_b[nt * 16 + l16];
    }

    // Issue all 3 tile loads (x, Wg, Wn chunks) for chunk c into stage c&1.
    auto issue_chunk = [&](int c) {
        float* base = (c & 1) ? stage1 : stage0;
        const int k0 = c * KCHUNK;
        tdm_load_tile_f32((unsigned)(size_t)base,
                          x + (size_t)tok0 * MDIM + k0,
                          MDIM, TOKENS, MDIM, KCHUNK, BLK_TOK);
        tdm_load_tile_f32((unsigned)(size_t)(base + X_FLOATS),
                          Wg_w + k0, MDIM, NEXP, MDIM, KCHUNK, NEXP);
        tdm_load_tile_f32((unsigned)(size_t)(base + X_FLOATS + W_FLOATS),
                          Wn_w + k0, MDIM, NEXP, MDIM, KCHUNK, NEXP);
    };

#ifdef HAVE_TDM
    if (wave == 0) issue_chunk(0);   // prime the pipe (TDM ignores EXEC, 1/wave)
#endif

    // ---------------- K loop over 8 chunks of 128, TDM double-buffered ------
    for (int c = 0; c < NCHUNKS; ++c) {
        float* xlds = ((c & 1) ? stage1 : stage0);
        float* glds = xlds + X_FLOATS;
        float* nlds = glds + W_FLOATS;

#ifdef HAVE_TDM
        if (wave == 0) __builtin_amdgcn_s_wait_tensorcnt(0); // chunk c landed
        __syncthreads();   // data visible to all waves; chunk c-1 consumed
        if (wave == 0 && c + 1 < NCHUNKS) issue_chunk(c + 1); // async prefetch
#else
        // Fallback: cooperative synchronous staging.
        __syncthreads();
        const int k0 = c * KCHUNK;
#pragma unroll
        for (int i = 0; i < 8; ++i) {
            int s = tid + i * 256, row = s >> 5, c4 = (s & 31) << 2;
            *(v4f*)(glds + row * LDSPAD + c4) = *(const v4f*)(Wg_w + (size_t)row * MDIM + k0 + c4);
            *(v4f*)(nlds + row * LDSPAD + c4) = *(const v4f*)(Wn_w + (size_t)row * MDIM + k0 + c4);
        }
#pragma unroll
        for (int i = 0; i < 16; ++i) {
            int s = tid + i * 256, row = s >> 5, c4 = (s & 31) << 2;
            *(v4f*)(xlds + row * LDSPAD + c4) = *(const v4f*)(x + (size_t)(tok0 + row) * MDIM + k0 + c4);
        }
        __syncthreads();
#endif

        // A fragment: 16x4 fp32 A -> lane l holds row M=l%16, K pair (hi*2,+1).
        const float* xa = xlds + (wave * 16 + l16) * LDSPAD + hi * 2;

#pragma unroll 4
        for (int kk = 0; kk < KCHUNK; kk += 4) {
            v2f a = *(const v2f*)(xa + kk);
#pragma unroll
            for (int nt = 0; nt < 4; ++nt) {
                const int brow = nt * 16 + l16;  // expert (N) for B frag
                v2f bgf = *(const v2f*)(glds + brow * LDSPAD + kk + hi * 2);
                v2f bnf = *(const v2f*)(nlds + brow * LDSPAD + kk + hi * 2);
                accG[nt] = __builtin_amdgcn_wmma_f32_16x16x4_f32(
                    false, a, false, bgf, (short)0, accG[nt], false, false);
                accN[nt] = __builtin_amdgcn_wmma_f32_16x16x4_f32(
                    false, a, false, bnf, (short)0, accN[nt], false, false);
            }
        }
    }

    // ---------------- Epilogue: bias, softplus, H, direct stores ------------
    // C/D layout: lane l, VGPR r -> element (M = r + 8*(l/16), N = l%16).
    // Re-stage H (16 x 64, stride 68) into this wave's private x region of the
    // last-used stage buffer (no TDM writes pending there).
    float* xlast = ((NCHUNKS - 1) & 1) ? stage1 : stage0;
    float* hbuf  = xlast + wave * 16 * LDSPAD;

#pragma unroll
    for (int nt = 0; nt < 4; ++nt) {
        const int e = nt * 16 + l16;
#pragma unroll
        for (int r = 0; r < 8; ++r) {
            const int trow  = r + hi * 8;
            const int token = tok0 + wave * 16 + trow;
            const size_t o  = (size_t)token * NEXP + e;

            float lg  = accG[nt][r] + bg[nt];
            float nl  = accN[nt][r] + bn[nt];
            // stable softplus: max(x,0) + log1p(exp(-|x|))
            float nsc = fmaxf(nl, 0.0f) + log1pf(expf(-fabsf(nl)));
            float h   = lg + noise[o] * nsc;

            lg_out[o] = lg;
            ns_out[o] = nsc;
            H_out[o]  = h;
            hbuf[trow * 68 + e] = h;
        }
    }
    __syncthreads();   // wave-internal LDS visibility

    // ---------------- top-2 over 64 experts ----------------
    // Lane pair (l16, l16+16) splits the 64 experts; merge via shfl_xor(16).
    float v0 = -__builtin_inff(), v1 = -__builtin_inff();
    int   i0 = 0, i1 = 0;
    {
        const float* hrow = hbuf + l16 * 68 + hi * 32;
#pragma unroll 8
        for (int j = 0; j < 32; ++j) {
            float v = hrow[j];
            int   e = hi * 32 + j;
            if (v > v0)      { v1 = v0; i1 = i0; v0 = v; i0 = e; }
            else if (v > v1) { v1 = v;  i1 = e; }
        }
    }
    float pv0 = __shfl_xor(v0, 16, 32);
    int   pi0 = __shfl_xor(i0, 16, 32);
    float pv1 = __shfl_xor(v1, 16, 32);
    int   pi1 = __shfl_xor(i1, 16, 32);
    if (pv0 > v0) {
        float tv = v0; int ti = i0;
        v0 = pv0; i0 = pi0;
        if (tv > pv1) { v1 = tv;  i1 = ti;  }
        else          { v1 = pv1; i1 = pi1; }
    } else if (pv0 > v1) {
        v1 = pv0; i1 = pi0;
    }

    if (hi == 0) {
        const int token = tok0 + wave * 16 + l16;
        float p0 = 1.0f / (1.0f + expf(v1 - v0));   // softmax over the pair
        float p1 = 1.0f - p0;
        idx_out[token * 2 + 0] = (float)i0;
        idx_out[token * 2 + 1] = (float)i1;
        float* grow = gates_out + (size_t)token * NEXP;
#pragma unroll
        for (int q = 0; q < 16; ++q) {
            v4f g = (v4f)0.0f;
            const int base = q * 4;
            if (i0 >= base && i0 < base + 4) g[i0 - base] = p0;
            if (i1 >= base && i1 < base + 4) g[i1 - base] = p1;
            *(v4f*)(grow + base) = g;
        }
    }
}

extern "C" void kernel_launch(void* const* d_in, const int* in_sizes, int n_in,
                              void* d_out, int out_size, void* d_ws, size_t ws_size,
                              hipStream_t stream) {
    const float* x     = (const float*)d_in[0];
    const float* noise = (const float*)d_in[1];
    const float* Wg_w  = (const float*)d_in[2];
    const float* Wg_b  = (const float*)d_in[3];
    const float* Wn_w  = (const float*)d_in[4];
    const float* Wn_b  = (const float*)d_in[5];

    float* out   = (float*)d_out;                       // outputs concat, return order:
    float* gates = out;                                 // [65536,64]
    float* H     = gates + (size_t)TOKENS * NEXP;       // [65536,64]
    float* idx   = H     + (size_t)TOKENS * NEXP;       // [65536,2] (as float)
    float* ns    = idx   + (size_t)TOKENS * 2;          // [65536,64]
    float* lg    = ns    + (size_t)TOKENS * NEXP;       // [65536,64]

    (void)hipFuncSetAttribute((const void*)noisy_topk_gate_kernel,
                              hipFuncAttributeMaxDynamicSharedMemorySize, SMEM_BYTES);

    dim3 grid(TOKENS / BLK_TOK);   // 512 workgroups
    dim3 block(256);               // 8 wave32 waves
    noisy_topk_gate_kernel<<<grid, block, SMEM_BYTES, stream>>>(
        x, noise, Wg_w, Wg_b, Wn_w, Wn_b, gates, H, idx, ns, lg);
}